// DistanceTransform_66683662238011
// MI455X (gfx1250) — compile-verified
//
#include <hip/hip_runtime.h>
#include <cstdint>

#define W     128
#define NPIX  (W * W)          // 16384 pixels per image
#define NT    1024             // 32 wave32 waves, one WGP per image
#define NITER 128              // ceil(max(H,W) / (k//2)) = 128
#define HPAR  0.35f

typedef __attribute__((ext_vector_type(4))) float f32x4;

__global__ __launch_bounds__(NT, 1)
void dt_kernel(const float* __restrict__ img, float* __restrict__ out)
{
    // ping-pong boundary buffers: 2 x 64KB, fits easily in 320KB/WGP LDS
    __shared__ __align__(16) float bufA[NPIX];
    __shared__ __align__(16) float bufB[NPIX];

    const int tid  = threadIdx.x;
    const int y    = tid >> 3;          // row 0..127
    const int x0   = (tid & 7) << 4;    // col start: 0,16,...,112 (64B aligned)
    const int base = y * W + x0;

    const uint64_t gimg = (uint64_t)(uintptr_t)(img + (size_t)blockIdx.x * NPIX);

    // ---- CDNA5 async bulk copy: global -> LDS (ASYNCcnt-tracked, no VGPR staging) ----
    {
        uint32_t ldsA = (uint32_t)(uintptr_t)(__attribute__((address_space(3))) float*)&bufA[base];
        uint32_t goff = (uint32_t)base * 4u;
        #pragma unroll
        for (int i = 0; i < 4; ++i) {
            uint32_t l = ldsA + 16u * (uint32_t)i;
            uint32_t g = goff + 16u * (uint32_t)i;
            asm volatile("global_load_async_to_lds_b128 %0, %1, %2"
                         :: "v"(l), "v"(g), "s"(gimg) : "memory");
        }
    }
    asm volatile("s_wait_asynccnt 0" ::: "memory");
    __syncthreads();

    // kernel weights: exp(-dist/h), dist in {0,1,sqrt2}; center weight is 1
    const float we  = __expf(-1.0f / HPAR);
    const float wcr = __expf(-1.4142135623730951f / HPAR);

    // branch-free replicate padding: precomputed clamped halo indices
    const int ym = (y == 0)       ? 0     : y - 1;
    const int yp = (y == W - 1)   ? W - 1 : y + 1;
    const int xl = (x0 == 0)      ? 0     : x0 - 1;
    const int xr = (x0 == W - 16) ? W - 1 : x0 + 16;

    float acc[16];
    #pragma unroll
    for (int i = 0; i < 16; ++i) acc[i] = 0.0f;

    #pragma unroll 2              // lets (it & 1) fold -> no pointer selects
    for (int it = 0; it < NITER; ++it) {
        const float* cur = (it & 1) ? bufB : bufA;
        float*       nxt = (it & 1) ? bufA : bufB;

        // m = middle row (18 cols incl. halo), u = top+bottom rows pre-summed
        float m[18], u[18];
        m[0]  = cur[y * W + xl];
        m[17] = cur[y * W + xr];
        u[0]  = cur[ym * W + xl] + cur[yp * W + xl];
        u[17] = cur[ym * W + xr] + cur[yp * W + xr];
        {
            const f32x4* pm = (const f32x4*)&cur[y  * W + x0];   // ds_load_b128
            const f32x4* pt = (const f32x4*)&cur[ym * W + x0];
            const f32x4* pb = (const f32x4*)&cur[yp * W + x0];
            #pragma unroll
            for (int j = 0; j < 4; ++j) {
                f32x4 vm = pm[j], vt = pt[j], vb = pb[j];
                m[1+4*j+0]=vm.x; m[1+4*j+1]=vm.y; m[1+4*j+2]=vm.z; m[1+4*j+3]=vm.w;
                u[1+4*j+0]=vt.x+vb.x; u[1+4*j+1]=vt.y+vb.y;
                u[1+4*j+2]=vt.z+vb.z; u[1+4*j+3]=vt.w+vb.w;
            }
        }

        const float offs = (float)((it * 3) >> 1);   // torch: i * kernel_size // 2

        float nb[16];
        #pragma unroll
        for (int i = 0; i < 16; ++i) {
            // conv = m[x] + we*(m[x-1]+m[x+1]+u[x]) + wc*(u[x-1]+u[x+1])
            float s1   = m[i] + m[i + 2] + u[i + 1];
            float conv = fmaf(we, s1, m[i + 1]);
            conv       = fmaf(wcr, u[i] + u[i + 2], conv);
            float cdt  = -HPAR * __logf(conv);       // v_log_f32 (TRANS, co-executes)
            bool  msk  = cdt > 0.0f;
            acc[i] += msk ? (offs + cdt) : 0.0f;
            nb[i]  = msk ? 1.0f : m[i + 1];
        }

        // ping-pong write: no pre-store barrier needed, one s_barrier per iteration
        {
            f32x4* q = (f32x4*)&nxt[base];           // ds_store_b128
            #pragma unroll
            for (int j = 0; j < 4; ++j) {
                f32x4 v; v.x=nb[4*j+0]; v.y=nb[4*j+1]; v.z=nb[4*j+2]; v.w=nb[4*j+3];
                q[j] = v;
            }
        }
        __syncthreads();
    }

    // single HBM write of the result
    float* gout = out + (size_t)blockIdx.x * NPIX + base;
    #pragma unroll
    for (int j = 0; j < 4; ++j) {
        f32x4 v; v.x=acc[4*j+0]; v.y=acc[4*j+1]; v.z=acc[4*j+2]; v.w=acc[4*j+3];
        ((f32x4*)gout)[j] = v;                        // global_store_b128
    }
}

extern "C" void kernel_launch(void* const* d_in, const int* in_sizes, int n_in,
                              void* d_out, int out_size, void* d_ws, size_t ws_size,
                              hipStream_t stream)
{
    (void)n_in; (void)out_size; (void)d_ws; (void)ws_size;
    const float* img = (const float*)d_in[0];
    float* out = (float*)d_out;
    const int B = in_sizes[0] / NPIX;   // 16 images -> 16 workgroups, one per WGP
    dt_kernel<<<B, NT, 0, stream>>>(img, out);
}